// LengthRegulator_37245956391152
// MI455X (gfx1250) — compile-verified
//
#include <hip/hip_runtime.h>
#include <hip/hip_bf16.h>
#include <stdint.h>

#ifndef __has_builtin
#define __has_builtin(x) 0
#endif

#if __has_builtin(__builtin_amdgcn_global_load_async_to_lds_b128) && \
    __has_builtin(__builtin_amdgcn_global_store_async_from_lds_b128) && \
    __has_builtin(__builtin_amdgcn_s_wait_asynccnt)
#define LR_HAVE_ASYNC 1
#else
#define LR_HAVE_ASYNC 0
#endif

namespace {
constexpr int B_SZ        = 32;
constexpr int L_ENC       = 512;
constexpr int L_DEC       = 2048;
constexpr int E_DIM       = 256;                 // floats per row (1 KB)
constexpr int WAVES_PER_BLOCK = 8;               // 256 threads, wave32
constexpr int ROWS_TOTAL  = B_SZ * L_DEC;        // 65536 output rows
constexpr size_t AMAP_BYTES = (size_t)ROWS_TOTAL * sizeof(unsigned long long); // 512 KB
}

// 16-byte int vector, as required by the async b128 builtins
typedef int v4i __attribute__((vector_size(16)));
typedef __attribute__((address_space(1))) v4i GVec;   // global
typedef __attribute__((address_space(3))) v4i LVec;   // LDS

__device__ __forceinline__ GVec* to_global_v(const void* p) {
    // flat and global address values are identical for global memory
    return (GVec*)(uintptr_t)p;
}
__device__ __forceinline__ LVec* to_lds_v(void* p) {
    // ISA aperture rule: the LDS address is the low 32 bits of the flat address
    return (LVec*)(uint32_t)(uintptr_t)p;
}

// ---------------------------------------------------------------------------
// Hot kernel: one output row (1 KB) per wave32, straight-line code.
// amap[r] holds the absolute source address for row r (encoder row or the
// zero row). The row is staged through LDS with async b128 transfers.
// ---------------------------------------------------------------------------
__global__ __launch_bounds__(32 * WAVES_PER_BLOCK) void lr_gather(
    const unsigned long long* __restrict__ amap,  // (B*L_DEC) src addresses
    float* __restrict__ out)                      // (B, L_DEC, E)
{
    __shared__ float tile[WAVES_PER_BLOCK][E_DIM];   // 8 KB
    const int wib  = threadIdx.x >> 5;
    const int lane = threadIdx.x & 31;
    const int r    = blockIdx.x * WAVES_PER_BLOCK + wib;   // flat (b,d)

    const unsigned long long sa = amap[r];   // uniform across the wave
    float* dst = out + (size_t)r * E_DIM;

#if LR_HAVE_ASYNC
    // each lane moves 32 B: two b128 transactions at byte offsets 0 / 16
    GVec* gs = (GVec*)(uintptr_t)(sa + (unsigned)(lane * 32));
    LVec* ls = to_lds_v(&tile[wib][lane * 8]);
    __builtin_amdgcn_global_load_async_to_lds_b128(gs, ls, 0, 0);
    __builtin_amdgcn_global_load_async_to_lds_b128(gs, ls, 16, 0);
    __builtin_amdgcn_s_wait_asynccnt(0);   // async LDS accesses are unordered
    GVec* gd = to_global_v(dst) + lane * 2;
    __builtin_amdgcn_global_store_async_from_lds_b128(gd, ls, 0, 0);
    __builtin_amdgcn_global_store_async_from_lds_b128(gd, ls, 16, 0);
    // S_ENDPGM performs an implicit wait-idle; stores drain before exit.
#else
    const float4* s4 = (const float4*)(sa + (unsigned)(lane * 32));
    float4*       d4 = (float4*)(dst + lane * 8);
    d4[0] = s4[0];
    d4[1] = s4[1];
#endif
}

// ---------------------------------------------------------------------------
// Kernel A: per-batch exclusive scan of durations; emits, for every decoder
// row, the absolute address of its source row (encoder row l with
// start[l] <= d < start[l]+dur[l], else the shared zero row).
// Also (re)writes the zero row each call (ws is not re-poisoned).
// ---------------------------------------------------------------------------
__global__ __launch_bounds__(L_ENC) void lr_build_map(
    const int*   __restrict__ durs,   // (B, L_ENC) int32 per harness contract
    const float* __restrict__ enc,    // (B, L_ENC, E) base for addresses
    float* __restrict__ zrow,         // (E) in workspace
    unsigned long long* __restrict__ amap)  // (B*L_DEC) in workspace
{
    __shared__ int s[L_ENC];
    __shared__ int smap[L_DEC];
    const int b = blockIdx.x;
    const int t = threadIdx.x;

    if (t < E_DIM) zrow[t] = 0.0f;  // all blocks write identical zeros: benign

    int d = durs[b * L_ENC + t];
    s[t] = d;
    __syncthreads();

    // Hillis-Steele inclusive scan over 512 elements (9 steps)
    for (int off = 1; off < L_ENC; off <<= 1) {
        int v = s[t];
        if (t >= off) v += s[t - off];
        __syncthreads();
        s[t] = v;
        __syncthreads();
    }
    const int start = s[t] - d;     // exclusive prefix = this row's first slot

    for (int j = t; j < L_DEC; j += L_ENC) smap[j] = -1;
    __syncthreads();
    for (int k = 0; k < d; ++k) {   // intervals are disjoint -> no conflicts
        const int p = start + k;
        if (p < L_DEC) smap[p] = t;
    }
    __syncthreads();
    for (int j = t; j < L_DEC; j += L_ENC) {
        const int idx = smap[j];
        const unsigned long long a = (idx >= 0)
            ? (unsigned long long)(uintptr_t)(enc + ((size_t)b * L_ENC + idx) * E_DIM)
            : (unsigned long long)(uintptr_t)zrow;
        amap[b * L_DEC + j] = a;
    }
}

// ---------------------------------------------------------------------------
extern "C" void kernel_launch(void* const* d_in, const int* in_sizes, int n_in,
                              void* d_out, int out_size, void* d_ws, size_t ws_size,
                              hipStream_t stream) {
    const float* enc  = (const float*)d_in[0];   // encoder_output f32
    const int*   durs = (const int*)d_in[1];     // durations (int)
    // d_in[2] = decoder_max_seq_len scalar; shapes fixed by the reference.
    unsigned long long* amap = (unsigned long long*)d_ws;   // 512 KB
    float* zrow = (float*)((char*)d_ws + AMAP_BYTES);       // 1 KB zero row
    float* out  = (float*)d_out;

    lr_build_map<<<B_SZ, L_ENC, 0, stream>>>(durs, enc, zrow, amap);
    lr_gather<<<ROWS_TOTAL / WAVES_PER_BLOCK, 32 * WAVES_PER_BLOCK, 0, stream>>>(
        amap, out);
}